// MinMaxLayer_77352361001485
// MI455X (gfx1250) — compile-verified
//
#include <hip/hip_runtime.h>
#include <stdint.h>

#define N_ROWS 32768
#define L_LEN  1000
#define R_POOL 5
#define WAVES_PER_BLOCK 8
#define ROW_PAD 1024           // floats per row slice in LDS (padded, 4KB)
#define GRANULES 250           // ceil(1000/4) 16-byte granules of real data

__global__ __launch_bounds__(256)
void minmax_pool_sort_kernel(const float* __restrict__ x,
                             const int* __restrict__ lengths,
                             float* __restrict__ out) {
    __shared__ float smem[WAVES_PER_BLOCK * ROW_PAD];

    const int lane = threadIdx.x & 31;
    const int wave = threadIdx.x >> 5;
    const int row  = blockIdx.x * WAVES_PER_BLOCK + wave;
    if (row >= N_ROWS) return;

    const float* rowp = x + (size_t)row * L_LEN;
    float* srow = &smem[wave * ROW_PAD];

    // LDS byte offset of this wave's slice: low 32 bits of the flat pointer
    // are the group-segment-relative LDS address.
    unsigned lds_base = (unsigned)(unsigned long long)(uintptr_t)srow;

    // Stage the row into LDS with the CDNA5 async data mover.
    // 8 iterations x 32 lanes x 16B = 4KB of LDS slots; source granules
    // clamped to the last in-bounds granule (slots >= 1000 are never read).
    #pragma unroll
    for (int k = 0; k < 8; ++k) {
        int g  = k * 32 + lane;                 // granule index 0..255
        int gc = g < GRANULES ? g : (GRANULES - 1);
        unsigned long long gaddr =
            (unsigned long long)(uintptr_t)(rowp + gc * 4);
        unsigned laddr = lds_base + (unsigned)g * 16u;
        asm volatile("global_load_async_to_lds_b128 %0, %1, off"
                     :: "v"(laddr), "v"(gaddr)
                     : "memory");
    }
    asm volatile("s_wait_asynccnt 0" ::: "memory");

    const int len = lengths[row];   // in [R_POOL, L_LEN] -> every window non-empty

    float v[2 * R_POOL];

    #pragma unroll
    for (int j = 0; j < R_POOL; ++j) {
        const int s = (j * len) / R_POOL;
        const int e = ((j + 1) * len + (R_POOL - 1)) / R_POOL;  // ceil
        float m = -__builtin_inff();
        float n =  __builtin_inff();
        for (int t = s + lane; t < e; t += 32) {
            float val = srow[t];
            m = fmaxf(m, val);
            n = fminf(n, val);
        }
        // wave32 butterfly reduction
        #pragma unroll
        for (int off = 16; off >= 1; off >>= 1) {
            m = fmaxf(m, __shfl_xor(m, off, 32));
            n = fminf(n, __shfl_xor(n, off, 32));
        }
        v[j]          = m;   // maxs[0..4]
        v[R_POOL + j] = n;   // mins[0..4]
    }

    // Ascending sort of 10 uniform values: fully-unrolled compare-exchange
    // network (constant indices keep v[] in registers).
    #pragma unroll
    for (int i = 0; i < 2 * R_POOL; ++i) {
        #pragma unroll
        for (int j2 = 0; j2 < 2 * R_POOL - 1 - i; ++j2) {
            float lo = fminf(v[j2], v[j2 + 1]);
            float hi = fmaxf(v[j2], v[j2 + 1]);
            v[j2]     = lo;
            v[j2 + 1] = hi;
        }
    }

    if (lane == 0) {
        float* orow = out + (size_t)row * (2 * R_POOL);
        #pragma unroll
        for (int k = 0; k < 2 * R_POOL; ++k) orow[k] = v[k];
    }
}

extern "C" void kernel_launch(void* const* d_in, const int* in_sizes, int n_in,
                              void* d_out, int out_size, void* d_ws, size_t ws_size,
                              hipStream_t stream) {
    (void)in_sizes; (void)n_in; (void)out_size; (void)d_ws; (void)ws_size;
    const float* x       = (const float*)d_in[0];
    const int*   lengths = (const int*)d_in[1];
    float*       out     = (float*)d_out;

    dim3 grid(N_ROWS / WAVES_PER_BLOCK);   // 4096 blocks
    dim3 block(32 * WAVES_PER_BLOCK);      // 256 threads = 8 wave32s
    minmax_pool_sort_kernel<<<grid, block, 0, stream>>>(x, lengths, out);
}